// HybridGNN_11003706212566
// MI455X (gfx1250) — compile-verified
//
#include <hip/hip_runtime.h>
#include <math.h>

// ---------------------------------------------------------------------------
// Problem constants (match reference)
// ---------------------------------------------------------------------------
#define NN 200000
#define EE 4000000
#define RR 4
#define DD 64
#define GG 64
#define CC 10

static_assert(NN % 16 == 0, "node tiling assumes N %% 16 == 0");

typedef __attribute__((ext_vector_type(2))) float v2f;
typedef __attribute__((ext_vector_type(8))) float v8f;

static inline int cdiv(int a, int b) { return (a + b - 1) / b; }

// ---------------------------------------------------------------------------
// h0 = op_emb[x_op] + cat_emb[x_cat]   (float4 per thread)
// ---------------------------------------------------------------------------
__global__ void embed_kernel(const int* __restrict__ xop, const int* __restrict__ xcat,
                             const float* __restrict__ ope, const float* __restrict__ cate,
                             float* __restrict__ h, int n) {
    int i = blockIdx.x * blockDim.x + threadIdx.x;      // over n*16 float4s
    if (i >= n * 16) return;
    int node = i >> 4, q = i & 15;
    const float4* a = (const float4*)(ope + (size_t)xop[node] * DD);
    const float4* b = (const float4*)(cate + (size_t)xcat[node] * DD);
    float4 va = a[q], vb = b[q];
    float4 r;
    r.x = va.x + vb.x; r.y = va.y + vb.y; r.z = va.z + vb.z; r.w = va.w + vb.w;
    ((float4*)(h + (size_t)node * DD))[q] = r;
}

// ---------------------------------------------------------------------------
// degree / per-(node,rel) counts
// ---------------------------------------------------------------------------
__global__ void count_deg_kernel(const int* __restrict__ tgt, const int* __restrict__ et,
                                 int* __restrict__ deg, int* __restrict__ cnt_nr, int e) {
    int i = blockIdx.x * blockDim.x + threadIdx.x;
    if (i >= e) return;
    int t = tgt[i];
    atomicAdd(&deg[t], 1);
    atomicAdd(&cnt_nr[t * RR + et[i]], 1);
}

__global__ void inv_cnt_kernel(const int* __restrict__ cnt_nr, float* __restrict__ invc, int n) {
    int i = blockIdx.x * blockDim.x + threadIdx.x;
    if (i >= n) return;
    int c = cnt_nr[i];
    invc[i] = 1.0f / (float)(c > 1 ? c : 1);
}

// ---------------------------------------------------------------------------
// exclusive scan of deg -> row_start (tile 1024)
// ---------------------------------------------------------------------------
#define SCAN_TILE 1024
__global__ void scan_partial_kernel(const int* __restrict__ deg, int* __restrict__ partials, int n) {
    __shared__ int sh[256];
    int base = blockIdx.x * SCAN_TILE;
    int sum = 0;
    for (int i = threadIdx.x; i < SCAN_TILE; i += 256) {
        int idx = base + i;
        sum += (idx < n) ? deg[idx] : 0;
    }
    sh[threadIdx.x] = sum; __syncthreads();
    for (int off = 128; off > 0; off >>= 1) {
        if (threadIdx.x < off) sh[threadIdx.x] += sh[threadIdx.x + off];
        __syncthreads();
    }
    if (threadIdx.x == 0) partials[blockIdx.x] = sh[0];
}

__global__ void scan_serial_kernel(int* __restrict__ partials, int nb) {
    int run = 0;
    for (int i = 0; i < nb; i++) { int t = partials[i]; partials[i] = run; run += t; }
}

__global__ void scan_apply_kernel(const int* __restrict__ deg, const int* __restrict__ partials,
                                  int* __restrict__ row_start, int n) {
    __shared__ int sh[256];
    int base = blockIdx.x * SCAN_TILE;
    int idx0 = base + threadIdx.x * 4;
    int v[4]; int s = 0;
    for (int j = 0; j < 4; j++) { int idx = idx0 + j; v[j] = (idx < n) ? deg[idx] : 0; s += v[j]; }
    sh[threadIdx.x] = s; __syncthreads();
    for (int off = 1; off < 256; off <<= 1) {
        int t = (threadIdx.x >= off) ? sh[threadIdx.x - off] : 0;
        __syncthreads();
        sh[threadIdx.x] += t;
        __syncthreads();
    }
    int excl = ((threadIdx.x > 0) ? sh[threadIdx.x - 1] : 0) + partials[blockIdx.x];
    for (int j = 0; j < 4; j++) {
        int idx = idx0 + j;
        if (idx < n) row_start[idx] = excl;
        excl += v[j];
    }
}

// ---------------------------------------------------------------------------
// CSR fill: packed = src | (etype << 28)
// ---------------------------------------------------------------------------
__global__ void fill_csr_kernel(const int* __restrict__ src, const int* __restrict__ tgt,
                                const int* __restrict__ et, const int* __restrict__ row_start,
                                int* __restrict__ cursor, int* __restrict__ packed, int e) {
    int i = blockIdx.x * blockDim.x + threadIdx.x;
    if (i >= e) return;
    int t = tgt[i];
    int pos = row_start[t] + atomicAdd(&cursor[t], 1);
    packed[pos] = src[i] | (et[i] << 28);
}

// ---------------------------------------------------------------------------
// Dense GEMMs via V_WMMA_F32_16X16X4_F32.
// Block = 160 threads (5 waves); blockIdx.x = 16-row node tile.
// Wave r<4 computes trans[r] = Htile @ W_r ; wave 4 computes root transform.
// A 16x4 layout: lanes 0-15 -> K=k0,k0+1 ; lanes 16-31 -> K=k0+2,k0+3 (M=lane&15)
// B 4x16 layout mirrored; C/D standard 8-VGPR layout.
// N is a multiple of 16 -> no row guards, straight-line stores.
// ---------------------------------------------------------------------------
#define HS 65   // padded LDS row stride (floats) to avoid bank conflicts
__global__ void rgcn_gemm_kernel(const float* __restrict__ h, const float* __restrict__ w_rel,
                                 const float* __restrict__ w_root, float* __restrict__ trans, int n) {
    __shared__ float Ht[16 * HS];
    int row0 = blockIdx.x * 16;
    for (int i = threadIdx.x; i < 16 * DD; i += blockDim.x) {
        int m = i >> 6, k = i & 63;
        Ht[m * HS + k] = h[(size_t)(row0 + m) * DD + k];
    }
    __syncthreads();

    int wave = threadIdx.x >> 5;      // 0..4
    int lane = threadIdx.x & 31;
    const float* W = (wave < 4) ? (w_rel + (size_t)wave * DD * DD) : w_root;
    float* out = trans + (size_t)wave * n * DD;
    int m = lane & 15, hi = lane >> 4;

    for (int nt = 0; nt < 4; nt++) {
        v8f c = {0.f, 0.f, 0.f, 0.f, 0.f, 0.f, 0.f, 0.f};
        int col = nt * 16 + m;
        const float* Wc = W + col;
        const float* Ha = Ht + m * HS + 2 * hi;
        for (int k0 = 0; k0 < DD; k0 += 4) {
            v2f a, b;
            a.x = Ha[k0];
            a.y = Ha[k0 + 1];
            b.x = Wc[(k0 + 2 * hi) * DD];
            b.y = Wc[(k0 + 2 * hi + 1) * DD];
            c = __builtin_amdgcn_wmma_f32_16x16x4_f32(false, a, false, b, (short)0, c,
                                                      false, false);
        }
        float* orow = out + (size_t)(row0 + 8 * hi) * DD + nt * 16 + m;
        #pragma unroll
        for (int i = 0; i < 8; i++) {
            orow[(size_t)i * DD] = c[i];
        }
    }
}

// ---------------------------------------------------------------------------
// Aggregation: one wave per target node; lane holds channels [2*lane, 2*lane+1].
// h_out[t] = (relu?)( root(t) + bias + sum_e invc[t,r_e] * trans[r_e][src_e] )
// 4-wide unroll with two independent accumulators: 4 outstanding random row
// loads per iteration instead of a serial load->fma chain. Prefetch the edge
// list stream ahead (global_prefetch_b8).
// ---------------------------------------------------------------------------
__global__ void aggregate_kernel(const float* __restrict__ trans, const int* __restrict__ row_start,
                                 const int* __restrict__ deg, const int* __restrict__ packed,
                                 const float* __restrict__ invc, const float* __restrict__ bias,
                                 float* __restrict__ hout, int n, int do_relu) {
    int wid = (blockIdx.x * blockDim.x + threadIdx.x) >> 5;
    if (wid >= n) return;
    int lane = threadIdx.x & 31;

    const float2 rootv = *(const float2*)(trans + ((size_t)4 * n + wid) * DD + lane * 2);
    const float2 bi = *(const float2*)(bias + lane * 2);
    float2 acc0; acc0.x = rootv.x + bi.x; acc0.y = rootv.y + bi.y;
    float2 acc1; acc1.x = 0.f; acc1.y = 0.f;

    const int s = row_start[wid], d = deg[wid];
    const float* invr = invc + wid * RR;

    int j = 0;
    for (; j + 4 <= d; j += 4) {
        if (j + 16 < d) __builtin_prefetch(&packed[s + j + 16], 0, 0);
        int p0 = packed[s + j + 0];
        int p1 = packed[s + j + 1];
        int p2 = packed[s + j + 2];
        int p3 = packed[s + j + 3];
        const float2 v0 = *(const float2*)(trans + ((size_t)(p0 >> 28) * n + (p0 & 0x0FFFFFFF)) * DD + lane * 2);
        const float2 v1 = *(const float2*)(trans + ((size_t)(p1 >> 28) * n + (p1 & 0x0FFFFFFF)) * DD + lane * 2);
        const float2 v2 = *(const float2*)(trans + ((size_t)(p2 >> 28) * n + (p2 & 0x0FFFFFFF)) * DD + lane * 2);
        const float2 v3 = *(const float2*)(trans + ((size_t)(p3 >> 28) * n + (p3 & 0x0FFFFFFF)) * DD + lane * 2);
        float w0 = invr[p0 >> 28];
        float w1 = invr[p1 >> 28];
        float w2 = invr[p2 >> 28];
        float w3 = invr[p3 >> 28];
        acc0.x += w0 * v0.x; acc0.y += w0 * v0.y;
        acc1.x += w1 * v1.x; acc1.y += w1 * v1.y;
        acc0.x += w2 * v2.x; acc0.y += w2 * v2.y;
        acc1.x += w3 * v3.x; acc1.y += w3 * v3.y;
    }
    for (; j < d; j++) {
        int p = packed[s + j];
        int src = p & 0x0FFFFFFF;
        int r = p >> 28;
        float w = invr[r];
        const float2 v = *(const float2*)(trans + ((size_t)r * n + src) * DD + lane * 2);
        acc0.x += w * v.x; acc0.y += w * v.y;
    }

    float2 acc; acc.x = acc0.x + acc1.x; acc.y = acc0.y + acc1.y;
    if (do_relu) { acc.x = fmaxf(acc.x, 0.f); acc.y = fmaxf(acc.y, 0.f); }
    *(float2*)(hout + (size_t)wid * DD + lane * 2) = acc;
}

// ---------------------------------------------------------------------------
// Global mean pool (atomic accumulate)
// ---------------------------------------------------------------------------
__global__ void pool_sum_kernel(const float* __restrict__ h, const int* __restrict__ batch,
                                float* __restrict__ psum, int* __restrict__ pcnt, int n) {
    int i = blockIdx.x * blockDim.x + threadIdx.x;   // over n*DD
    if (i >= n * DD) return;
    int node = i >> 6, d = i & 63;
    int g = batch[node];
    atomicAdd(&psum[g * DD + d], h[i]);
    if (d == 0) atomicAdd(&pcnt[g], 1);
}

// ---------------------------------------------------------------------------
// Head: mean, fc1+relu, fc2, log_softmax. One thread per graph (G=64).
// ---------------------------------------------------------------------------
__global__ void head_kernel(const float* __restrict__ psum, const int* __restrict__ pcnt,
                            const float* __restrict__ fc1w, const float* __restrict__ fc1b,
                            const float* __restrict__ fc2w, const float* __restrict__ fc2b,
                            float* __restrict__ out) {
    int g = threadIdx.x;
    if (g >= GG) return;
    float inv = 1.0f / fmaxf((float)pcnt[g], 1.0f);
    float p[DD];
    for (int d = 0; d < DD; d++) p[d] = psum[g * DD + d] * inv;
    float t[DD];
    for (int j = 0; j < DD; j++) {
        float s = fc1b[j];
        for (int d = 0; d < DD; d++) s += p[d] * fc1w[d * DD + j];
        t[j] = fmaxf(s, 0.f);
    }
    float lo[CC]; float mx = -1e30f;
    for (int c = 0; c < CC; c++) {
        float s = fc2b[c];
        for (int d = 0; d < DD; d++) s += t[d] * fc2w[d * CC + c];
        lo[c] = s;
        mx = fmaxf(mx, s);
    }
    float se = 0.f;
    for (int c = 0; c < CC; c++) se += expf(lo[c] - mx);
    float lse = mx + logf(se);
    for (int c = 0; c < CC; c++) out[g * CC + c] = lo[c] - lse;
}

// ---------------------------------------------------------------------------
// Host launch
// ---------------------------------------------------------------------------
extern "C" void kernel_launch(void* const* d_in, const int* in_sizes, int n_in,
                              void* d_out, int out_size, void* d_ws, size_t ws_size,
                              hipStream_t stream) {
    const int* x_op   = (const int*)d_in[0];
    const int* x_cat  = (const int*)d_in[1];
    const int* e_src  = (const int*)d_in[2];            // edge_index[0]
    const int* e_tgt  = ((const int*)d_in[2]) + EE;     // edge_index[1]
    const int* e_type = (const int*)d_in[3];
    const int* batch  = (const int*)d_in[4];
    const float* op_emb  = (const float*)d_in[5];
    const float* cat_emb = (const float*)d_in[6];
    const float* w_[3]   = {(const float*)d_in[7],  (const float*)d_in[10], (const float*)d_in[13]};
    const float* r_[3]   = {(const float*)d_in[8],  (const float*)d_in[11], (const float*)d_in[14]};
    const float* b_[3]   = {(const float*)d_in[9],  (const float*)d_in[12], (const float*)d_in[15]};
    const float* fc1w = (const float*)d_in[16];
    const float* fc1b = (const float*)d_in[17];
    const float* fc2w = (const float*)d_in[18];
    const float* fc2b = (const float*)d_in[19];
    float* out = (float*)d_out;

    // bump allocator over d_ws
    char* p = (char*)d_ws;
    auto alloc = [&](size_t bytes) {
        char* q = p;
        p += (bytes + 255) & ~(size_t)255;
        return (void*)q;
    };
    float* trans   = (float*)alloc((size_t)(RR + 1) * NN * DD * 4);  // 256 MB (4 rel + root)
    float* hA      = (float*)alloc((size_t)NN * DD * 4);             // 51.2 MB
    float* hB      = (float*)alloc((size_t)NN * DD * 4);             // 51.2 MB
    int*   cnt_nr  = (int*)alloc((size_t)NN * RR * 4);
    float* invc    = (float*)alloc((size_t)NN * RR * 4);
    int*   deg     = (int*)alloc((size_t)NN * 4);
    int*   rstart  = (int*)alloc((size_t)NN * 4);
    int*   cursor  = (int*)alloc((size_t)NN * 4);
    int*   packed  = (int*)alloc((size_t)EE * 4);
    int*   partials= (int*)alloc(1024);
    float* psum    = (float*)alloc((size_t)GG * DD * 4);
    int*   pcnt    = (int*)alloc((size_t)GG * 4);
    (void)ws_size; (void)n_in; (void)in_sizes; (void)out_size;

    hipMemsetAsync(cnt_nr, 0, (size_t)NN * RR * 4, stream);
    hipMemsetAsync(deg,    0, (size_t)NN * 4, stream);
    hipMemsetAsync(cursor, 0, (size_t)NN * 4, stream);
    hipMemsetAsync(psum,   0, (size_t)GG * DD * 4, stream);
    hipMemsetAsync(pcnt,   0, (size_t)GG * 4, stream);

    // h0
    embed_kernel<<<cdiv(NN * 16, 256), 256, 0, stream>>>(x_op, x_cat, op_emb, cat_emb, hA, NN);

    // degree + per-(node,rel) counts, inverse
    count_deg_kernel<<<cdiv(EE, 256), 256, 0, stream>>>(e_tgt, e_type, deg, cnt_nr, EE);
    inv_cnt_kernel<<<cdiv(NN * RR, 256), 256, 0, stream>>>(cnt_nr, invc, NN * RR);

    // exclusive scan deg -> rstart
    int nb = cdiv(NN, SCAN_TILE);
    scan_partial_kernel<<<nb, 256, 0, stream>>>(deg, partials, NN);
    scan_serial_kernel<<<1, 1, 0, stream>>>(partials, nb);
    scan_apply_kernel<<<nb, 256, 0, stream>>>(deg, partials, rstart, NN);

    // CSR fill
    fill_csr_kernel<<<cdiv(EE, 256), 256, 0, stream>>>(e_src, e_tgt, e_type, rstart, cursor, packed, EE);

    // 3 RGCN layers
    float* hc = hA;
    float* hn = hB;
    int tiles = NN / 16;
    for (int L = 0; L < 3; L++) {
        rgcn_gemm_kernel<<<tiles, 160, 0, stream>>>(hc, w_[L], r_[L], trans, NN);
        aggregate_kernel<<<cdiv(NN * 32, 256), 256, 0, stream>>>(
            trans, rstart, deg, packed, invc, b_[L], hn, NN, (L < 2) ? 1 : 0);
        float* tmp = hc; hc = hn; hn = tmp;
    }

    // pool + head
    pool_sum_kernel<<<cdiv(NN * DD, 256), 256, 0, stream>>>(hc, batch, psum, pcnt, NN);
    head_kernel<<<1, 64, 0, stream>>>(psum, pcnt, fc1w, fc1b, fc2w, fc2b, out);
}